// DeformableHeadAttention_34961033790029
// MI455X (gfx1250) — compile-verified
//
#include <hip/hip_runtime.h>
#include <math.h>

// ---------------------------------------------------------------------------
// Deformable head attention, MI455X (gfx1250), wave32.
// GEMMs: V_WMMA_F32_16X16X4_F32 (exact fp32), W staged to LDS via
// gfx1250 async global->LDS copies, B fragments served from LDS.
// Each wave computes a 16x32 output tile (2 accumulators, shared A fragment).
// ---------------------------------------------------------------------------

typedef __attribute__((ext_vector_type(2))) float v2f;
typedef __attribute__((ext_vector_type(8))) float v8f;

#define C_DIM 128   // model dim / GEMM K
#define NQ    16384 // H*W

// Async copy of one 16-byte chunk global -> LDS (ASYNCcnt-tracked).
__device__ __forceinline__ void async_copy_b128(unsigned lds_byte_off,
                                                const void* gaddr) {
  asm volatile("global_load_async_to_lds_b128 %0, %1, off"
               :
               : "v"(lds_byte_off), "v"(gaddr)
               : "memory");
}
__device__ __forceinline__ void wait_asynccnt0() {
  asm volatile("s_wait_asynccnt 0x0" ::: "memory");
}

// ---------------------------------------------------------------------------
// GEMM: Out[n, j] = sum_k A[n,k]*W[k,j] + bias[j],  K fixed = 128.
// Wave computes 16 rows x 32 cols = two 16x16 tiles, 32 k-steps of f32 WMMA.
// A fragment (16x4): lane l -> row = l&15, K = kbase + 2*(l>>4) + {0,1}
// B fragment (4x16): lane l -> col = l&15, K = kbase + 2*(l>>4) + {0,1}
// D (16x16): VGPR i, lane l -> row = i + 8*(l>>4), col = l&15
// ---------------------------------------------------------------------------
template <int COUT>
__global__ __launch_bounds__(256) void gemm_bias_wmma_f32(
    const float* __restrict__ A, const float* __restrict__ W,
    const float* __restrict__ bias, float* __restrict__ Out, int nrows) {
  __shared__ float sW[C_DIM * COUT]; // exact row-major copy of W (<= 64KB)

  // ---- stage W into LDS with async b128 copies --------------------------
  {
    const unsigned ldsBase = (unsigned)(uintptr_t)(&sW[0]); // low 32b = LDS off
    constexpr int NCHUNK = (C_DIM * COUT) / 4; // 16B chunks
    const char* gw = (const char*)W;
    for (int ci = threadIdx.x; ci < NCHUNK; ci += 256) {
      async_copy_b128(ldsBase + (unsigned)ci * 16u, gw + (size_t)ci * 16);
    }
    wait_asynccnt0();
    __syncthreads();
  }

  constexpr int CG = COUT / 32; // column groups of 32
  const int wave = threadIdx.x >> 5;
  const int lane = threadIdx.x & 31;
  const int tile = blockIdx.x * 8 + wave;
  const int rowTile = tile / CG;
  const int cg = tile - rowTile * CG;
  const int row0 = rowTile << 4;
  if (row0 >= nrows) return;

  const int r = lane & 15;            // A row within tile / B col within tile
  const int khalf = (lane >> 4) << 1; // 0 or 2
  const int col0 = cg * 32 + r;       // first column this lane serves

  const float* __restrict__ arow = A + (size_t)(row0 + r) * C_DIM + khalf;
  const float* __restrict__ wl = &sW[(size_t)khalf * COUT + col0];

  v8f acc0 = {};
  v8f acc1 = {};
#pragma unroll
  for (int k = 0; k < C_DIM; k += 4) {
    v2f afrag = *(const v2f*)(arow + k); // K = k+khalf, k+khalf+1 (8B aligned)
    v2f b0, b1;
    b0.x = wl[k * COUT];
    b0.y = wl[(k + 1) * COUT];
    b1.x = wl[k * COUT + 16];
    b1.y = wl[(k + 1) * COUT + 16];
    acc0 = __builtin_amdgcn_wmma_f32_16x16x4_f32(false, afrag, false, b0,
                                                 (short)0, acc0, false, false);
    acc1 = __builtin_amdgcn_wmma_f32_16x16x4_f32(false, afrag, false, b1,
                                                 (short)0, acc1, false, false);
  }

  const float bv0 = bias[col0];
  const float bv1 = bias[col0 + 16];
  const int lhalf = lane >> 4;
#pragma unroll
  for (int i = 0; i < 8; ++i) {
    const size_t rowOff = (size_t)(row0 + i + 8 * lhalf) * COUT;
    Out[rowOff + col0] = acc0[i] + bv0;
    Out[rowOff + col0 + 16] = acc1[i] + bv1;
  }
}

// ---------------------------------------------------------------------------
// Softmax over contiguous groups of 8 (per query, per head over L*K).
// ---------------------------------------------------------------------------
__global__ __launch_bounds__(256) void softmax8_kernel(float* __restrict__ a,
                                                       int ngroups) {
  const int t = blockIdx.x * blockDim.x + threadIdx.x;
  if (t >= ngroups) return;
  float* p = a + (size_t)t * 8;
  float4 v0 = ((const float4*)p)[0];
  float4 v1 = ((const float4*)p)[1];
  float mx = fmaxf(fmaxf(fmaxf(v0.x, v0.y), fmaxf(v0.z, v0.w)),
                   fmaxf(fmaxf(v1.x, v1.y), fmaxf(v1.z, v1.w)));
  v0.x = __expf(v0.x - mx); v0.y = __expf(v0.y - mx);
  v0.z = __expf(v0.z - mx); v0.w = __expf(v0.w - mx);
  v1.x = __expf(v1.x - mx); v1.y = __expf(v1.y - mx);
  v1.z = __expf(v1.z - mx); v1.w = __expf(v1.w - mx);
  const float inv = 1.0f / (v0.x + v0.y + v0.z + v0.w + v1.x + v1.y + v1.z + v1.w);
  v0.x *= inv; v0.y *= inv; v0.z *= inv; v0.w *= inv;
  v1.x *= inv; v1.y *= inv; v1.z *= inv; v1.w *= inv;
  ((float4*)p)[0] = v0;
  ((float4*)p)[1] = v1;
}

// ---------------------------------------------------------------------------
// Faithful sampling + aggregation, replicating the reference's scrambled
// permute/view. For output (n = b*M+m, q, c'):
//   k   = q >> 12
//   w_s = (q >> 5) & 127
//   h_s = (q & 31) * 4 + (c' >> 2)
//   l   = (c' >> 1) & 1
//   c   = 8*(c' & 1) + s
//   out[n,q,c'] = sum_s bilinear(Wx_l[n, c], coords(n,l,k,h_s,w_s)) * An[n,q,s]
// phi uses p_q batch (n % B) — faithful to jnp.tile(phi,(M,1,1,1)).
// Wx kept in [B, h, w, M*C_v] layout => 16 contiguous channels per tap.
// ---------------------------------------------------------------------------
__global__ __launch_bounds__(256) void deform_sample_kernel(
    const float* __restrict__ deltas,   // [B, H, W, 128] cols = ((m*2+l)*4+k)*2+xy
    const float* __restrict__ asoft,    // [B, H, W, 64]  cols = m*8 + s
    const float* __restrict__ pq,       // [B, H, W, 2]
    const float* __restrict__ img_lo,   // [B, 64, 64, 128]
    const float* __restrict__ img_hi,   // [B, 128, 128, 128]
    float* __restrict__ val)            // [B, H, W, 128] cols = m*16 + c'
{
  const int t = blockIdx.x * blockDim.x + threadIdx.x;
  if (t >= 16 * NQ) return;
  const int n = t >> 14;     // b*M + m
  const int q = t & (NQ - 1);
  const int b = n >> 3;
  const int m = n & 7;
  const int kpt = q >> 12;          // sample point index
  const int w_s = (q >> 5) & 127;   // sample query col
  const int h_base = (q & 31) << 2; // sample query row base
  const int hq = q >> 7;            // output row
  const int wq = q & 127;           // output col

  // attention weights for this (n, q): An[s], s = l*K + k
  const float4* ap =
      (const float4*)(asoft + ((size_t)b * NQ + q) * 64 + m * 8);
  const float4 anA = ap[0];
  const float4 anB = ap[1];
  const float An[8] = {anA.x, anA.y, anA.z, anA.w, anB.x, anB.y, anB.z, anB.w};

  float out16[16];

#pragma unroll
  for (int hoff = 0; hoff < 4; ++hoff) {
    const int h_s = h_base + hoff;
    const size_t qs = (size_t)h_s * 128 + w_s;
    // faithful quirk: phi batch index = n % B
    const float px = pq[(((size_t)(n & 1)) * NQ + qs) * 2 + 0];
    const float py = pq[(((size_t)(n & 1)) * NQ + qs) * 2 + 1];
    const float* drow =
        deltas + ((size_t)b * NQ + qs) * 128 + (m * 16 + kpt * 2);

#pragma unroll
    for (int lbit = 0; lbit < 2; ++lbit) {
      const int lw = lbit ? 128 : 64;
      const int lh = lw;
      const float dx = drow[lbit * 8 + 0];
      const float dy = drow[lbit * 8 + 1];
      // coords then grid_sample index math (align_corners=False):
      // ix = cx * w/(w-1) - 0.5
      const float cx = px * (float)(lw - 1) + dx;
      const float cy = py * (float)(lh - 1) + dy;
      const float ix = cx * ((float)lw / (float)(lw - 1)) - 0.5f;
      const float iy = cy * ((float)lh / (float)(lh - 1)) - 0.5f;
      const float xf = floorf(ix);
      const float yf = floorf(iy);
      const int x0 = (int)xf;
      const int y0 = (int)yf;
      const float fx1 = ix - xf, fy1 = iy - yf;
      const float fx0 = 1.0f - fx1, fy0 = 1.0f - fy1;

      float4 s0 = {0.f, 0.f, 0.f, 0.f};
      float4 s1 = s0, s2 = s0, s3 = s0;
      const float* __restrict__ img = lbit ? img_hi : img_lo;

      const int xs[4] = {x0, x0 + 1, x0, x0 + 1};
      const int ys[4] = {y0, y0, y0 + 1, y0 + 1};
      const float wt[4] = {fx0 * fy0, fx1 * fy0, fx0 * fy1, fx1 * fy1};
#pragma unroll
      for (int tap = 0; tap < 4; ++tap) {
        if (xs[tap] >= 0 && xs[tap] < lw && ys[tap] >= 0 && ys[tap] < lh) {
          const float4* p =
              (const float4*)(img +
                              (((size_t)b * lh + ys[tap]) * lw + xs[tap]) * 128 +
                              m * 16);
          const float4 a0 = p[0], a1 = p[1], a2 = p[2], a3 = p[3];
          const float w = wt[tap];
          s0.x = fmaf(w, a0.x, s0.x); s0.y = fmaf(w, a0.y, s0.y);
          s0.z = fmaf(w, a0.z, s0.z); s0.w = fmaf(w, a0.w, s0.w);
          s1.x = fmaf(w, a1.x, s1.x); s1.y = fmaf(w, a1.y, s1.y);
          s1.z = fmaf(w, a1.z, s1.z); s1.w = fmaf(w, a1.w, s1.w);
          s2.x = fmaf(w, a2.x, s2.x); s2.y = fmaf(w, a2.y, s2.y);
          s2.z = fmaf(w, a2.z, s2.z); s2.w = fmaf(w, a2.w, s2.w);
          s3.x = fmaf(w, a3.x, s3.x); s3.y = fmaf(w, a3.y, s3.y);
          s3.z = fmaf(w, a3.z, s3.z); s3.w = fmaf(w, a3.w, s3.w);
        }
      }
      // channels 0..7 -> parity 0 output, channels 8..15 -> parity 1
      const float o0 = s0.x * An[0] + s0.y * An[1] + s0.z * An[2] + s0.w * An[3] +
                       s1.x * An[4] + s1.y * An[5] + s1.z * An[6] + s1.w * An[7];
      const float o1 = s2.x * An[0] + s2.y * An[1] + s2.z * An[2] + s2.w * An[3] +
                       s3.x * An[4] + s3.y * An[5] + s3.z * An[6] + s3.w * An[7];
      out16[hoff * 4 + lbit * 2 + 0] = o0;
      out16[hoff * 4 + lbit * 2 + 1] = o1;
    }
  }

  float* vp = val + (((size_t)b * NQ) + (size_t)hq * 128 + wq) * 128 + m * 16;
  ((float4*)vp)[0] = make_float4(out16[0], out16[1], out16[2], out16[3]);
  ((float4*)vp)[1] = make_float4(out16[4], out16[5], out16[6], out16[7]);
  ((float4*)vp)[2] = make_float4(out16[8], out16[9], out16[10], out16[11]);
  ((float4*)vp)[3] = make_float4(out16[12], out16[13], out16[14], out16[15]);
}

// ---------------------------------------------------------------------------
// Launcher
// ---------------------------------------------------------------------------
extern "C" void kernel_launch(void* const* d_in, const int* in_sizes, int n_in,
                              void* d_out, int out_size, void* d_ws,
                              size_t ws_size, hipStream_t stream) {
  (void)in_sizes; (void)n_in; (void)out_size; (void)ws_size;
  const float* z_q = (const float*)d_in[0];
  const float* x0  = (const float*)d_in[1];
  const float* x1  = (const float*)d_in[2];
  const float* p_q = (const float*)d_in[3];
  const float* Wq  = (const float*)d_in[4];
  const float* bq  = (const float*)d_in[5];
  const float* Wd  = (const float*)d_in[6];
  const float* bd  = (const float*)d_in[7];
  const float* Wa  = (const float*)d_in[8];
  const float* ba  = (const float*)d_in[9];
  const float* Wp  = (const float*)d_in[10];
  const float* bp  = (const float*)d_in[11];
  const float* Wm  = (const float*)d_in[12];
  const float* bm  = (const float*)d_in[13];
  float* out = (float*)d_out;

  // workspace layout (floats): total 19,922,944 floats = ~79.7 MB
  float* ws   = (float*)d_ws;
  float* zq   = ws;                 // 32768*128
  float* dlt  = zq + 4194304;       // 32768*128
  float* alog = dlt + 4194304;      // 32768*64
  float* wxlo = alog + 2097152;     // 8192*128  (level 0: 2x64x64)
  float* wxhi = wxlo + 1048576;     // 32768*128 (level 1: 2x128x128)
  float* val  = wxhi + 4194304;     // 32768*128

  const dim3 blk(256);
  // zq = z_q @ Wq + bq : 2048 row tiles x 4 col groups = 8192 waves
  gemm_bias_wmma_f32<128><<<1024, blk, 0, stream>>>(z_q, Wq, bq, zq, 32768);
  // deltas = zq @ Wd + bd
  gemm_bias_wmma_f32<128><<<1024, blk, 0, stream>>>(zq, Wd, bd, dlt, 32768);
  // attention logits = zq @ Wa + ba : 2048 x 2 = 4096 waves
  gemm_bias_wmma_f32<64><<<512, blk, 0, stream>>>(zq, Wa, ba, alog, 32768);
  // Wx per level = x_l @ Wp + bp
  gemm_bias_wmma_f32<128><<<256, blk, 0, stream>>>(x0, Wp, bp, wxlo, 8192);
  gemm_bias_wmma_f32<128><<<1024, blk, 0, stream>>>(x1, Wp, bp, wxhi, 32768);
  // softmax over L*K=8 per (query, head)
  softmax8_kernel<<<1024, blk, 0, stream>>>(alog, 262144);
  // bilinear sampling + scrambled aggregation (faithful)
  deform_sample_kernel<<<1024, blk, 0, stream>>>(dlt, alog, p_q, wxlo, wxhi,
                                                 val);
  // out = val @ Wm + bm
  gemm_bias_wmma_f32<128><<<1024, blk, 0, stream>>>(val, Wm, bm, out, 32768);
}